// PointNetSetAbstraction_73065983639805
// MI455X (gfx1250) — compile-verified
//
#include <hip/hip_runtime.h>

#define NB    8
#define NPTS  8192
#define SPTS  2048   // NPOINT
#define KSAMP 32
#define DPTS  32     // input point-feature channels
#define CIN   35     // 3 + DPTS
#define CINP  36     // padded to multiple of 4
#define H1    64
#define H2    64
#define H3    128
#define LDW   68     // LDS row stride (padded)

typedef __attribute__((ext_vector_type(2))) float v2f;
typedef __attribute__((ext_vector_type(8))) float v8f;

__device__ inline v8f wmma4(v2f a, v2f b, v8f c) {
  // D(16x16,f32) = A(16x4,f32) * B(4x16,f32) + C
  return __builtin_amdgcn_wmma_f32_16x16x4_f32(
      /*neg_a=*/false, a, /*neg_b=*/false, b,
      /*c_mod=*/(short)0, c, /*reuse_a=*/false, /*reuse_b=*/false);
}

// ---------------------------------------------------------------------------
// Kernel A: farthest point sampling. One block per batch, 1024 threads.
// Points + running min-distances live in registers (8 per thread).
// Emits new_xyz in both (B,3,S) layout (d_out part 1) and (B,S,3) (workspace).
// ---------------------------------------------------------------------------
__global__ __launch_bounds__(1024)
void fps_kernel(const float* __restrict__ xyz,
                float* __restrict__ out_newxyz,   // (B,3,S)
                float* __restrict__ nxw) {        // (B,S,3)
  const int b   = blockIdx.x;
  const int tid = threadIdx.x;
  const float* xb = xyz + (size_t)b * 3 * NPTS;

  float px[8], py[8], pz[8], dist[8];
#pragma unroll
  for (int i = 0; i < 8; ++i) {
    int n = tid + i * 1024;
    px[i] = xb[n];
    py[i] = xb[NPTS + n];
    pz[i] = xb[2 * NPTS + n];
    dist[i] = 1e10f;
  }

  __shared__ float sval[32];
  __shared__ int   sidx[32];
  __shared__ int   sfar;
  __shared__ float c3[3];
  if (tid == 0) { c3[0] = px[0]; c3[1] = py[0]; c3[2] = pz[0]; sfar = 0; }
  __syncthreads();

  const int lane = tid & 31, wid = tid >> 5;

  for (int it = 0; it < SPTS; ++it) {
    // record current farthest point's coordinates
    if (tid < 3) {
      float v = c3[tid];
      out_newxyz[(size_t)b * 3 * SPTS + (size_t)tid * SPTS + it] = v;
      nxw[((size_t)b * SPTS + it) * 3 + tid] = v;
    }
    float cx = c3[0], cy = c3[1], cz = c3[2];

    // update min-distances, local argmax (ties -> smaller index)
    float bv = -1.0f; int bi = 0;
#pragma unroll
    for (int i = 0; i < 8; ++i) {
      float dx = px[i] - cx, dy = py[i] - cy, dz = pz[i] - cz;
      float d = dx * dx + dy * dy + dz * dz;
      d = fminf(dist[i], d);
      dist[i] = d;
      int n = tid + i * 1024;
      if (d > bv || (d == bv && n < bi)) { bv = d; bi = n; }
    }
    // wave32 shuffle argmax
#pragma unroll
    for (int off = 16; off > 0; off >>= 1) {
      float ov = __shfl_xor(bv, off, 32);
      int   oi = __shfl_xor(bi, off, 32);
      if (ov > bv || (ov == bv && oi < bi)) { bv = ov; bi = oi; }
    }
    if (lane == 0) { sval[wid] = bv; sidx[wid] = bi; }
    __syncthreads();
    if (tid < 32) {
      bv = sval[tid]; bi = sidx[tid];
#pragma unroll
      for (int off = 16; off > 0; off >>= 1) {
        float ov = __shfl_xor(bv, off, 32);
        int   oi = __shfl_xor(bi, off, 32);
        if (ov > bv || (ov == bv && oi < bi)) { bv = ov; bi = oi; }
      }
      if (tid == 0) sfar = bi;
    }
    __syncthreads();
    int nf = sfar;
    if ((nf & 1023) == tid) {         // owner publishes next centroid coords
      int i = nf >> 10;
      c3[0] = px[i]; c3[1] = py[i]; c3[2] = pz[i];
    }
    __syncthreads();
  }
}

// ---------------------------------------------------------------------------
// Kernel B: ball query. One wave32 per query; ballot + prefix popcount keeps
// the first KSAMP in-radius indices in ascending order (matches sort-then-take).
// ---------------------------------------------------------------------------
__global__ __launch_bounds__(256)
void ballquery_kernel(const float* __restrict__ xyz,
                      const float* __restrict__ nxw,  // (B,S,3)
                      int* __restrict__ idxp) {       // (B,S,KSAMP)
  const int gw   = (int)((blockIdx.x * blockDim.x + threadIdx.x) >> 5);
  const int lane = threadIdx.x & 31;
  if (gw >= NB * SPTS) return;
  const int b = gw / SPTS;
  const float* xb = xyz + (size_t)b * 3 * NPTS;
  const float cx = nxw[gw * 3 + 0], cy = nxw[gw * 3 + 1], cz = nxw[gw * 3 + 2];

  int cnt = 0, firstIdx = 0;
  bool haveFirst = false;
  for (int n0 = 0; n0 < NPTS && cnt < KSAMP; n0 += 32) {
    int n = n0 + lane;
    float dx = xb[n] - cx, dy = xb[NPTS + n] - cy, dz = xb[2 * NPTS + n] - cz;
    float d = dx * dx + dy * dy + dz * dz;
    bool within = (d <= 0.25f);                       // RADIUS^2
    unsigned mask = (unsigned)__ballot(within);       // wave32: low 32 bits
    if (!haveFirst && mask) { firstIdx = n0 + __builtin_ctz(mask); haveFirst = true; }
    int pos = cnt + __builtin_popcount(mask & ((1u << lane) - 1u));
    if (within && pos < KSAMP) idxp[(size_t)gw * KSAMP + pos] = n;
    cnt += __builtin_popcount(mask);
  }
  if (cnt < KSAMP && lane >= cnt)                     // pad with first index
    idxp[(size_t)gw * KSAMP + lane] = firstIdx;
}

// ---------------------------------------------------------------------------
// Kernel C: gather + 3-layer MLP (WMMA f32 16x16x4) + max over samples.
// 128 threads = 4 waves; one wave per query (32 samples = 2 M-tiles).
// Activations staged in a per-wave LDS tile 32 x LDW.
// ---------------------------------------------------------------------------
__global__ __launch_bounds__(128)
void group_mlp_kernel(const float* __restrict__ xyz, const float* __restrict__ pts,
                      const float* __restrict__ nxw, const int* __restrict__ idxp,
                      const float* __restrict__ w1, const float* __restrict__ b1,
                      const float* __restrict__ w2, const float* __restrict__ b2,
                      const float* __restrict__ w3, const float* __restrict__ b3,
                      float* __restrict__ outp) {     // (B,128,S)
  __shared__ float buf[4][KSAMP][LDW];
  const int wid  = threadIdx.x >> 5;
  const int lane = threadIdx.x & 31;
  const int q = blockIdx.x * 4 + wid;                 // global query id
  const int b = q / SPTS, s = q % SPTS;
  float (*T)[LDW] = buf[wid];

  // ---- build feat rows: [grouped_xyz - center | grouped_points], pad col 35
  const int gi = idxp[(size_t)q * KSAMP + lane];
  const float* xb = xyz + (size_t)b * 3 * NPTS;
  const float* pb = pts + (size_t)b * DPTS * NPTS;
  const float cx = nxw[q * 3 + 0], cy = nxw[q * 3 + 1], cz = nxw[q * 3 + 2];
  T[lane][0] = xb[gi] - cx;
  T[lane][1] = xb[NPTS + gi] - cy;
  T[lane][2] = xb[2 * NPTS + gi] - cz;
#pragma unroll
  for (int d = 0; d < DPTS; ++d) T[lane][3 + d] = pb[(size_t)d * NPTS + gi];
  T[lane][35] = 0.0f;
  __syncthreads();

  const int nlo = lane & 15;            // M (A) / N (B,C) coordinate
  const int kb  = (lane >> 4) << 1;     // K sub-offset for A/B frags

  // ---------------- layer 1: 36(pad) -> 64 ----------------
  v8f acc1[4][2];
#pragma unroll
  for (int nt = 0; nt < 4; ++nt) { acc1[nt][0] = (v8f){}; acc1[nt][1] = (v8f){}; }
  for (int k0 = 0; k0 < CINP; k0 += 4) {
    v2f a0, a1;
    a0.x = T[nlo][k0 + kb];      a0.y = T[nlo][k0 + kb + 1];
    a1.x = T[16 + nlo][k0 + kb]; a1.y = T[16 + nlo][k0 + kb + 1];
    int c0 = k0 + kb;
#pragma unroll
    for (int nt = 0; nt < 4; ++nt) {
      int r = nt * 16 + nlo;
      v2f bf;
      bf.x = (c0 < CIN)     ? w1[r * CIN + c0]     : 0.0f;
      bf.y = (c0 + 1 < CIN) ? w1[r * CIN + c0 + 1] : 0.0f;
      acc1[nt][0] = wmma4(a0, bf, acc1[nt][0]);
      acc1[nt][1] = wmma4(a1, bf, acc1[nt][1]);
    }
  }
  __syncthreads();
#pragma unroll
  for (int nt = 0; nt < 4; ++nt) {
    float bc = b1[nt * 16 + nlo];
#pragma unroll
    for (int mt = 0; mt < 2; ++mt)
#pragma unroll
      for (int i = 0; i < 8; ++i) {
        int row = mt * 16 + i + ((lane >> 4) << 3);
        float v = acc1[nt][mt][i] + bc;
        T[row][nt * 16 + nlo] = v > 0.0f ? v : 0.0f;
      }
  }
  __syncthreads();

  // ---------------- layer 2: 64 -> 64 ----------------
  v8f acc2[4][2];
#pragma unroll
  for (int nt = 0; nt < 4; ++nt) { acc2[nt][0] = (v8f){}; acc2[nt][1] = (v8f){}; }
  for (int k0 = 0; k0 < H1; k0 += 4) {
    v2f a0, a1;
    a0.x = T[nlo][k0 + kb];      a0.y = T[nlo][k0 + kb + 1];
    a1.x = T[16 + nlo][k0 + kb]; a1.y = T[16 + nlo][k0 + kb + 1];
    int c0 = k0 + kb;
#pragma unroll
    for (int nt = 0; nt < 4; ++nt) {
      int r = nt * 16 + nlo;
      v2f bf;
      bf.x = w2[r * H1 + c0];
      bf.y = w2[r * H1 + c0 + 1];
      acc2[nt][0] = wmma4(a0, bf, acc2[nt][0]);
      acc2[nt][1] = wmma4(a1, bf, acc2[nt][1]);
    }
  }
  __syncthreads();
#pragma unroll
  for (int nt = 0; nt < 4; ++nt) {
    float bc = b2[nt * 16 + nlo];
#pragma unroll
    for (int mt = 0; mt < 2; ++mt)
#pragma unroll
      for (int i = 0; i < 8; ++i) {
        int row = mt * 16 + i + ((lane >> 4) << 3);
        float v = acc2[nt][mt][i] + bc;
        T[row][nt * 16 + nlo] = v > 0.0f ? v : 0.0f;
      }
  }
  __syncthreads();

  // ---------------- layer 3: 64 -> 128, fused max over 32 samples ----------
  for (int nt = 0; nt < 8; ++nt) {
    v8f c0a = (v8f){}, c1a = (v8f){};
    for (int k0 = 0; k0 < H2; k0 += 4) {
      v2f a0, a1, bf;
      a0.x = T[nlo][k0 + kb];      a0.y = T[nlo][k0 + kb + 1];
      a1.x = T[16 + nlo][k0 + kb]; a1.y = T[16 + nlo][k0 + kb + 1];
      int c0 = k0 + kb;
      int r = nt * 16 + nlo;
      bf.x = w3[r * H2 + c0];
      bf.y = w3[r * H2 + c0 + 1];
      c0a = wmma4(a0, bf, c0a);
      c1a = wmma4(a1, bf, c1a);
    }
    float bc = b3[nt * 16 + nlo];
    float m = 0.0f;                          // relu outputs are >= 0
#pragma unroll
    for (int i = 0; i < 8; ++i) {
      float v0 = c0a[i] + bc; v0 = v0 > 0.0f ? v0 : 0.0f;
      float v1 = c1a[i] + bc; v1 = v1 > 0.0f ? v1 : 0.0f;
      m = fmaxf(m, fmaxf(v0, v1));
    }
    m = fmaxf(m, __shfl_xor(m, 16, 32));     // combine lane halves (rows)
    if (lane < 16)
      outp[(size_t)b * H3 * SPTS + (size_t)(nt * 16 + lane) * SPTS + s] = m;
  }
}

// ---------------------------------------------------------------------------
extern "C" void kernel_launch(void* const* d_in, const int* in_sizes, int n_in,
                              void* d_out, int out_size, void* d_ws, size_t ws_size,
                              hipStream_t stream) {
  const float* xyz = (const float*)d_in[0];   // (B,3,N)
  const float* pts = (const float*)d_in[1];   // (B,32,N)
  const float* w1  = (const float*)d_in[2];
  const float* b1  = (const float*)d_in[3];
  const float* w2  = (const float*)d_in[4];
  const float* b2  = (const float*)d_in[5];
  const float* w3  = (const float*)d_in[6];
  const float* b3  = (const float*)d_in[7];

  float* out      = (float*)d_out;                      // new_xyz (B,3,S) first
  float* out_feat = out + (size_t)NB * 3 * SPTS;        // then (B,128,S)

  float* nxw  = (float*)d_ws;                           // (B,S,3)
  int*   idxp = (int*)((float*)d_ws + (size_t)NB * SPTS * 3);  // (B,S,32)

  fps_kernel<<<NB, 1024, 0, stream>>>(xyz, out, nxw);
  ballquery_kernel<<<(NB * SPTS) / 8, 256, 0, stream>>>(xyz, nxw, idxp);
  group_mlp_kernel<<<(NB * SPTS) / 4, 128, 0, stream>>>(
      xyz, pts, nxw, idxp, w1, b1, w2, b2, w3, b3, out_feat);
}